// FilterDetections_16698832846859
// MI455X (gfx1250) — compile-verified
//
#include <hip/hip_runtime.h>
#include <hip/hip_bf16.h>
#include <stdint.h>

#define NUM_CLASSES 16
#define MAX_DET     100
#define SCORE_THR   0.01f
#define NMS_THR     0.5f
#define NEG_SENT    (-1e30f)
#define TILE        1024
#define NMS_BLOCK   1024

// ---------------------------------------------------------------------------
// Kernel 1: transpose classification [N,16] -> thresholded scores [16][N]
//           and prefill candidate arrays with (-1, -1).
// ---------------------------------------------------------------------------
__global__ void fd_init_kernel(const float* __restrict__ cls,
                               float* __restrict__ scores_t,
                               float* __restrict__ cand_sc,
                               int*   __restrict__ cand_idx,
                               int N) {
    int i = blockIdx.x * blockDim.x + threadIdx.x;
    if (i < NUM_CLASSES * MAX_DET) { cand_sc[i] = -1.0f; cand_idx[i] = -1; }
    if (i < N) {
        #pragma unroll
        for (int c = 0; c < NUM_CLASSES; ++c) {
            float v = cls[(size_t)i * NUM_CLASSES + c];
            scores_t[(size_t)c * N + i] = (v > SCORE_THR) ? v : NEG_SENT;
        }
    }
}

// ---------------------------------------------------------------------------
// Async global->LDS staging of one tile of boxes (one float4 box per thread).
// Per-wave: one global_load_async_to_lds_b128 covering 32 lanes x 16B,
// tracked with ASYNCcnt (in-order completion) -- no block barrier needed
// since each thread consumes exactly the LDS slot it staged.
// ---------------------------------------------------------------------------
__device__ __forceinline__ void async_stage_tile(const float* __restrict__ boxes,
                                                 int N, int t, int buf, int tid,
                                                 uint32_t ldsBase) {
    int i = t * TILE + tid;
    if (i >= N) i = N - 1;                       // clamp: keep EXEC full, dup loads harmless
    uint64_t ga = (uint64_t)(uintptr_t)(boxes + (size_t)i * 4);
    uint32_t la = ldsBase + (uint32_t)((buf * TILE + tid) * 16);
#if defined(__HIP_DEVICE_COMPILE__)
    asm volatile("global_load_async_to_lds_b128 %0, %1, off"
                 :: "v"(la), "v"(ga) : "memory");
#endif
}

// ---------------------------------------------------------------------------
// Kernel 2: per-class greedy NMS. One block (1024 thr = 32 wave32) per class.
// 100 rounds: block-wide vectorized argmax (lowest index wins ties, matching
// jnp.argmax), then an IoU suppression sweep with double-buffered async box
// staging synchronized purely via s_wait_asynccnt.
// ---------------------------------------------------------------------------
__launch_bounds__(NMS_BLOCK)
__global__ void fd_nms_kernel(const float* __restrict__ boxes,
                              float* __restrict__ scores_t,
                              float* __restrict__ cand_sc,
                              int*   __restrict__ cand_idx,
                              int N) {
    const int c    = blockIdx.x;
    const int tid  = threadIdx.x;
    const int lane = tid & 31;
    const int wave = tid >> 5;
    float* sc = scores_t + (size_t)c * N;
    const float4* sc4 = reinterpret_cast<const float4*>(sc);

    __shared__ float s_boxes[2][TILE * 4];        // 32 KB double buffer
    __shared__ float s_rv[NMS_BLOCK / 32];
    __shared__ int   s_ri[NMS_BLOCK / 32];
    __shared__ float s_bv;
    __shared__ int   s_bi;
    __shared__ float s_win[4];

    const int tiles = (N + TILE - 1) / TILE;
    const int N4    = N >> 2;                     // full float4 groups
    const uint32_t ldsBase = (uint32_t)(uintptr_t)(&s_boxes[0][0]);

    for (int r = 0; r < MAX_DET; ++r) {
        // ---------- pass 1: vectorized argmax over suppressed score array ----------
        float bv = -__builtin_inff();
        int   bi = 0x7fffffff;
        for (int g = tid; g < N4; g += NMS_BLOCK) {
            float4 v = sc4[g];
            int base = g << 2;
            // first-max within the quad (lower index wins ties)
            float v01 = (v.x >= v.y) ? v.x : v.y;
            int   i01 = (v.x >= v.y) ? base : base + 1;
            float v23 = (v.z >= v.w) ? v.z : v.w;
            int   i23 = (v.z >= v.w) ? base + 2 : base + 3;
            float vq  = (v01 >= v23) ? v01 : v23;
            int   iq  = (v01 >= v23) ? i01 : i23;
            if (vq > bv) { bv = vq; bi = iq; }    // strict >: earlier group wins ties
        }
        // tail elements [N4*4, N)
        {
            int i = (N4 << 2) + tid;
            if (i < N) {
                float v = sc[i];
                if (v > bv || (v == bv && i < bi)) { bv = v; bi = i; }
            }
        }
        #pragma unroll
        for (int o = 16; o > 0; o >>= 1) {
            float ov = __shfl_down(bv, o);
            int   oi = __shfl_down(bi, o);
            if (ov > bv || (ov == bv && oi < bi)) { bv = ov; bi = oi; }
        }
        if (lane == 0) { s_rv[wave] = bv; s_ri[wave] = bi; }
        __syncthreads();
        if (wave == 0) {
            bv = s_rv[lane]; bi = s_ri[lane];
            #pragma unroll
            for (int o = 16; o > 0; o >>= 1) {
                float ov = __shfl_down(bv, o);
                int   oi = __shfl_down(bi, o);
                if (ov > bv || (ov == bv && oi < bi)) { bv = ov; bi = oi; }
            }
            if (lane == 0) {
                s_bv = bv; s_bi = bi;
                if (bv > SCORE_THR) {
                    cand_sc[c * MAX_DET + r]  = bv;
                    cand_idx[c * MAX_DET + r] = bi;
                    s_win[0] = boxes[(size_t)bi * 4 + 0];
                    s_win[1] = boxes[(size_t)bi * 4 + 1];
                    s_win[2] = boxes[(size_t)bi * 4 + 2];
                    s_win[3] = boxes[(size_t)bi * 4 + 3];
                }
            }
        }
        __syncthreads();
        if (!(s_bv > SCORE_THR)) break;           // exact early exit: scores only decrease

        const float wx1 = s_win[0], wy1 = s_win[1], wx2 = s_win[2], wy2 = s_win[3];
        const float warea = (wx2 - wx1) * (wy2 - wy1);

        // ---------- pass 2: suppress with double-buffered async staging ----------
        async_stage_tile(boxes, N, 0, 0, tid, ldsBase);
        for (int t = 0; t < tiles; ++t) {
            const int buf = t & 1;
            if (t + 1 < tiles) {
                async_stage_tile(boxes, N, t + 1, buf ^ 1, tid, ldsBase);
#if defined(__HIP_DEVICE_COMPILE__)
                asm volatile("s_wait_asynccnt 1" ::: "memory");   // oldest tile landed
#endif
            } else {
#if defined(__HIP_DEVICE_COMPILE__)
                asm volatile("s_wait_asynccnt 0" ::: "memory");   // drain
#endif
            }
            const int i = t * TILE + tid;
            if (i < N) {
                float si = sc[i];
                if (si != NEG_SENT) {             // skip already-suppressed
                    const float* bp = &s_boxes[buf][tid * 4];
                    float x1 = bp[0], y1 = bp[1], x2 = bp[2], y2 = bp[3];
                    float ix1 = fmaxf(wx1, x1), iy1 = fmaxf(wy1, y1);
                    float ix2 = fminf(wx2, x2), iy2 = fminf(wy2, y2);
                    float inter = fmaxf(ix2 - ix1, 0.0f) * fmaxf(iy2 - iy1, 0.0f);
                    float area  = (x2 - x1) * (y2 - y1);
                    float iou   = inter / (warea + area - inter);
                    if (iou > NMS_THR) sc[i] = NEG_SENT;   // winner self-suppresses (IoU=1)
                }
            }
        }
        __threadfence_block();   // WGP-scope visibility is sufficient: block-private data
        __syncthreads();
    }
}

// ---------------------------------------------------------------------------
// Kernel 3: merge 16*100 candidates -> global top-100 (sorted desc, ties by
// lower flat index, matching lax.top_k), gather outputs.
// Output layout (floats): boxes[100*4] | scores[100] | labels[100] |
//                         rotation[100*3] | translation[100*3]  = 1200
// ---------------------------------------------------------------------------
__global__ void fd_merge_kernel(const float* __restrict__ boxes,
                                const float* __restrict__ rot,
                                const float* __restrict__ trans,
                                const float* __restrict__ cand_sc,
                                const int*   __restrict__ cand_idx,
                                float* __restrict__ out) {
    const int tid  = threadIdx.x;                 // 128 threads = 4 waves
    const int lane = tid & 31;
    const int wave = tid >> 5;
    const int TOT  = NUM_CLASSES * MAX_DET;       // 1600

    __shared__ float s_sc[NUM_CLASSES * MAX_DET];
    __shared__ int   s_ix[NUM_CLASSES * MAX_DET];
    __shared__ float s_rv[4];
    __shared__ int   s_ri[4];

    for (int j = tid; j < TOT; j += 128) { s_sc[j] = cand_sc[j]; s_ix[j] = cand_idx[j]; }
    __syncthreads();

    for (int r = 0; r < MAX_DET; ++r) {
        float bv = -__builtin_inff();
        int   bj = 0x7fffffff;
        for (int j = tid; j < TOT; j += 128) {
            float v = s_sc[j];
            if (v > bv) { bv = v; bj = j; }
        }
        #pragma unroll
        for (int o = 16; o > 0; o >>= 1) {
            float ov = __shfl_down(bv, o);
            int   oj = __shfl_down(bj, o);
            if (ov > bv || (ov == bv && oj < bj)) { bv = ov; bj = oj; }
        }
        if (lane == 0) { s_rv[wave] = bv; s_ri[wave] = bj; }
        __syncthreads();
        if (tid == 0) {
            bv = s_rv[0]; bj = s_ri[0];
            #pragma unroll
            for (int w = 1; w < 4; ++w) {
                if (s_rv[w] > bv || (s_rv[w] == bv && s_ri[w] < bj)) { bv = s_rv[w]; bj = s_ri[w]; }
            }
            const bool valid = bv > SCORE_THR;
            int sel = valid ? s_ix[bj] : 0;
            if (sel < 0) sel = 0;
            const int label = bj / MAX_DET;
            #pragma unroll
            for (int k = 0; k < 4; ++k)
                out[r * 4 + k] = valid ? boxes[(size_t)sel * 4 + k] : -1.0f;
            out[400 + r] = valid ? bv : -1.0f;
            out[500 + r] = valid ? (float)label : -1.0f;
            #pragma unroll
            for (int k = 0; k < 3; ++k) {
                out[600 + r * 3 + k] = valid ? rot[(size_t)sel * 3 + k]   : -1.0f;
                out[900 + r * 3 + k] = valid ? trans[(size_t)sel * 3 + k] : -1.0f;
            }
            s_sc[bj] = NEG_SENT;                  // consume
        }
        __syncthreads();
    }
}

// ---------------------------------------------------------------------------
extern "C" void kernel_launch(void* const* d_in, const int* in_sizes, int n_in,
                              void* d_out, int out_size, void* d_ws, size_t ws_size,
                              hipStream_t stream) {
    const float* boxes = (const float*)d_in[0];   // [N,4]
    const float* cls   = (const float*)d_in[1];   // [N,16]
    const float* rot   = (const float*)d_in[2];   // [N,3]
    const float* trans = (const float*)d_in[3];   // [N,3]
    const int N = in_sizes[0] / 4;

    // workspace layout
    float* scores_t = (float*)d_ws;                               // 16*N floats
    float* cand_sc  = scores_t + (size_t)NUM_CLASSES * N;         // 1600 floats
    int*   cand_idx = (int*)(cand_sc + NUM_CLASSES * MAX_DET);    // 1600 ints
    float* out      = (float*)d_out;                              // 1200 floats

    int total = (N > NUM_CLASSES * MAX_DET) ? N : NUM_CLASSES * MAX_DET;
    int blocks = (total + 255) / 256;
    fd_init_kernel<<<blocks, 256, 0, stream>>>(cls, scores_t, cand_sc, cand_idx, N);
    fd_nms_kernel<<<NUM_CLASSES, NMS_BLOCK, 0, stream>>>(boxes, scores_t, cand_sc, cand_idx, N);
    fd_merge_kernel<<<1, 128, 0, stream>>>(boxes, rot, trans, cand_sc, cand_idx, out);
}